// PatchWiseProcessingLayer_16569983828349
// MI455X (gfx1250) — compile-verified
//
#include <hip/hip_runtime.h>
#include <hip/hip_bf16.h>
#include <math.h>

// Patch-wise normalization, (16,512,512,3) fp32, 32x32 patches stride 16.
// Pass 1: per-patch (invstd, mean*invstd) via WMMA-assisted wave reduction.
// Pass 2: elementwise out = (x*A - B)/count  (A,B gathered from <=4 covering patches).

typedef __attribute__((ext_vector_type(2))) float v2f;
typedef __attribute__((ext_vector_type(8))) float v8f;

#define NH 31
#define NW 31
#define IMG_H 512
#define IMG_W 512
#define CH 3
#define ROW_F4 (IMG_W * CH / 4)          // 384 float4 per image row
#define PATCH_ROW_F4 (32 * CH / 4)       // 24 float4 per patch row
#define PATCH_ELEMS (32 * 32 * CH)       // 3072

__global__ __launch_bounds__(256)
void patch_stats_kernel(const float* __restrict__ x, float2* __restrict__ stats) {
    const int blk = blockIdx.x;                 // (b*31 + ph)*31 + pw
    const int pw  = blk % NW;
    const int ph  = (blk / NW) % NH;
    const int b   = blk / (NH * NW);

    // Patch base: byte offset = 6144*rows + 192*pw -> 16B aligned, float4 legal.
    const float4* p4 = (const float4*)(x + (((size_t)b * IMG_H + (size_t)(ph * 16)) * IMG_W
                                            + (size_t)(pw * 16)) * CH);

    const int tid = threadIdx.x;
    float s = 0.0f, q = 0.0f;
    // 768 float4 / 256 threads = 3 global_load_b128 per thread, coalesced.
    #pragma unroll
    for (int i = 0; i < 3; ++i) {
        int f   = tid + i * 256;
        int row = f / PATCH_ROW_F4;
        int c4  = f - row * PATCH_ROW_F4;
        float4 v = p4[(size_t)row * ROW_F4 + c4];
        s += v.x + v.y + v.z + v.w;
        q = fmaf(v.x, v.x, q);
        q = fmaf(v.y, v.y, q);
        q = fmaf(v.z, v.z, q);
        q = fmaf(v.w, v.w, q);
    }

    // ---- wave32 reduction of (s,q) via V_WMMA_F32_16X16X4_F32 (exact fp32 FMAs) ----
    // A (16x4): VGPR0 = s -> A[lane&15, lane<16?0:2]; VGPR1 = q -> A[lane&15, lane<16?1:3]
    // B (4x16): selector; cols 0-7 take K in {0,2} (sums), cols 8-15 take K in {1,3} (sumsqs)
    const unsigned lane = (unsigned)tid & 31u;
    v2f A; A.x = s; A.y = q;
    const float selLo = ((lane & 15u) < 8u) ? 1.0f : 0.0f;
    v2f Bm; Bm.x = selLo; Bm.y = 1.0f - selLo;
    v8f D = {};
    D = __builtin_amdgcn_wmma_f32_16x16x4_f32(false, A, false, Bm, (short)0, D, false, false);
    // D[m, n<8] = s_m + s_{m+16};  D[m, n>=8] = q_m + q_{m+16}  (all 32 partials consumed)
    // Per-lane fold of the 8 D VGPRs: lanes 0-15 hold sum over M=0..7 of column (lane&15),
    // lanes 16-31 hold sum over M=8..15. So lanes {0,16} carry the S halves and
    // lanes {8,24} carry the Q halves -> write straight to LDS, no shuffle needed.
    float c = D[0] + D[1] + D[2] + D[3] + D[4] + D[5] + D[6] + D[7];

    __shared__ float sW[16], qW[16];
    const int wv = tid >> 5;                 // wave id (8 waves)
    const unsigned l15  = lane & 15u;
    const unsigned half = lane >> 4;
    if (l15 == 0u) sW[wv * 2 + half] = c;
    if (l15 == 8u) qW[wv * 2 + half] = c;
    __syncthreads();

    if (tid == 0) {
        float S = 0.0f, Q = 0.0f;
        #pragma unroll
        for (int w = 0; w < 16; ++w) { S += sW[w]; Q += qW[w]; }
        float mean = S * (1.0f / (float)PATCH_ELEMS);
        float var  = Q * (1.0f / (float)PATCH_ELEMS) - mean * mean;
        float inv  = 1.0f / sqrtf(var + 1e-6f);
        stats[blk] = make_float2(inv, mean * inv);
    }
}

__global__ __launch_bounds__(256)
void patch_apply_kernel(const float* __restrict__ x, const float2* __restrict__ stats,
                        float* __restrict__ out) {
    const int g    = blockIdx.x * 256 + threadIdx.x;  // 4-pixel group
    const int pix0 = g << 2;
    const int b    = pix0 >> 18;                      // / (512*512)
    const int rem  = pix0 & 262143;
    const int i    = rem >> 9;                        // row
    const int j0   = rem & 511;                       // first col (multiple of 4)

    const float4* X4 = (const float4*)x;
    float4*       O4 = (float4*)out;
    const size_t  base4 = (size_t)g * 3;              // 12 floats = 3 float4, 16B aligned
    const float4 a0 = X4[base4 + 0];
    const float4 a1 = X4[base4 + 1];
    const float4 a2 = X4[base4 + 2];

    // covering patch-row range for this image row
    const int plo = (i >= 31) ? ((i - 16) >> 4) : 0;
    const int phi = min(30, i >> 4);
    const float2* srow = stats + (size_t)b * (NH * NW);

    float sc[4], of[4];
    #pragma unroll
    for (int u = 0; u < 4; ++u) {
        const int j   = j0 + u;
        const int wlo = (j >= 31) ? ((j - 16) >> 4) : 0;
        const int whi = min(30, j >> 4);
        float As = 0.0f, Bs = 0.0f;
        for (int ph = plo; ph <= phi; ++ph) {
            for (int pw = wlo; pw <= whi; ++pw) {
                const float2 st = srow[ph * NW + pw];  // L1/L2 resident, heavy reuse
                As += st.x;                             // sum of invstd
                Bs += st.y;                             // sum of mean*invstd
            }
        }
        // count in {1,2,4} -> reciprocal multiply is exact
        const float rc = 1.0f / (float)((phi - plo + 1) * (whi - wlo + 1));
        sc[u] = As * rc;
        of[u] = Bs * rc;
    }

    float4 o0, o1, o2;
    o0.x = fmaf(a0.x, sc[0], -of[0]);
    o0.y = fmaf(a0.y, sc[0], -of[0]);
    o0.z = fmaf(a0.z, sc[0], -of[0]);
    o0.w = fmaf(a0.w, sc[1], -of[1]);
    o1.x = fmaf(a1.x, sc[1], -of[1]);
    o1.y = fmaf(a1.y, sc[1], -of[1]);
    o1.z = fmaf(a1.z, sc[2], -of[2]);
    o1.w = fmaf(a1.w, sc[2], -of[2]);
    o2.x = fmaf(a2.x, sc[2], -of[2]);
    o2.y = fmaf(a2.y, sc[3], -of[3]);
    o2.z = fmaf(a2.z, sc[3], -of[3]);
    o2.w = fmaf(a2.w, sc[3], -of[3]);
    O4[base4 + 0] = o0;
    O4[base4 + 1] = o1;
    O4[base4 + 2] = o2;
}

extern "C" void kernel_launch(void* const* d_in, const int* in_sizes, int n_in,
                              void* d_out, int out_size, void* d_ws, size_t ws_size,
                              hipStream_t stream) {
    (void)in_sizes; (void)n_in; (void)out_size; (void)ws_size;
    const float* x   = (const float*)d_in[0];
    float*       out = (float*)d_out;
    float2*      st  = (float2*)d_ws;   // 16*31*31 float2 = 123 KB

    // Pass 1: one workgroup (8 waves) per patch, 16*31*31 = 15376 patches.
    patch_stats_kernel<<<16 * NH * NW, 256, 0, stream>>>(x, st);
    // Pass 2: 16*512*512/4 = 1,048,576 threads -> 4096 blocks.
    patch_apply_kernel<<<4096, 256, 0, stream>>>(x, st, out);
}